// Top_HiCL_9612136808770
// MI455X (gfx1250) — compile-verified
//
#include <hip/hip_runtime.h>
#include <hip/hip_bf16.h>
#include <math.h>

#define DIM    256
#define NROWS  50000
#define NEDGE  1600000
#define BSEL   1024
#define NNEGS  32
#define INV_TEMP 5.0f

typedef __attribute__((ext_vector_type(2))) float v2f;
typedef __attribute__((ext_vector_type(8))) float v8f;

// ---------------------------------------------------------------------------
// CDNA5 WMMA: D = A(16x4 f32) * B(4x16 f32) + C(16x16 f32), one tile per wave32.
// A: lane m = lane&15, VGPR j holds K = k0 + 2*(lane>>4) + j. B symmetric.
// C/D: acc[v] is element (m0 + v + 8*(lane>>4), n0 + (lane&15)).
// ---------------------------------------------------------------------------
__device__ __forceinline__ v8f wmma_f32_16x16x4(v2f a, v2f b, v8f c) {
#if defined(__HIP_DEVICE_COMPILE__) && __has_builtin(__builtin_amdgcn_wmma_f32_16x16x4_f32)
  return __builtin_amdgcn_wmma_f32_16x16x4_f32(false, a, false, b, (short)0, c, false, false);
#else
  c[0] += a.x * b.x + a.y * b.y;  // host-pass placeholder only
  return c;
#endif
}

// ============================ CSR construction =============================
// counts[key[i]]++ (counts pre-zeroed)
__global__ void count_kernel(const int* __restrict__ keys, int* __restrict__ counts, int n) {
  int i = blockIdx.x * blockDim.x + threadIdx.x;
  if (i < n) atomicAdd(&counts[keys[i]], 1);
}

// per-256-block totals of counts
__global__ void scan_block_kernel(const int* __restrict__ counts, int* __restrict__ bsum, int n) {
  int t = threadIdx.x;
  int i = blockIdx.x * 256 + t;
  int v = (i < n) ? counts[i] : 0;
#pragma unroll
  for (int m = 1; m < 32; m <<= 1) v += __shfl_xor(v, m, 32);
  __shared__ int sh[8];
  int lane = t & 31, w = t >> 5;
  if (lane == 0) sh[w] = v;
  __syncthreads();
  if (t == 0) {
    int s = 0;
    for (int k = 0; k < 8; ++k) s += sh[k];
    bsum[blockIdx.x] = s;
  }
}

// serial exclusive scan of block sums (nb ~ 196; trivial)
__global__ void scan_spine_kernel(int* __restrict__ bsum, int nb) {
  if (threadIdx.x == 0 && blockIdx.x == 0) {
    int acc = 0;
    for (int k = 0; k < nb; ++k) { int t = bsum[k]; bsum[k] = acc; acc += t; }
  }
}

// offs[i] = exclusive_scan(counts)[i]; offs[n] = total
__global__ void scan_final_kernel(const int* __restrict__ counts, const int* __restrict__ bsum,
                                  int* __restrict__ offs, int n) {
  __shared__ int sh[256];
  int t = threadIdx.x;
  int i = blockIdx.x * 256 + t;
  int v = (i < n) ? counts[i] : 0;
  sh[t] = v;
  __syncthreads();
  for (int off = 1; off < 256; off <<= 1) {
    int x = (t >= off) ? sh[t - off] : 0;
    __syncthreads();
    sh[t] += x;
    __syncthreads();
  }
  int incl = sh[t];
  int base = bsum[blockIdx.x];
  if (i < n) offs[i] = base + incl - v;
  if (i == n - 1) offs[n] = base + incl;
}

// perm scatter: pos = cursor[key[i]]++; perm[pos] = i
__global__ void fill_perm_kernel(const int* __restrict__ keys, int* __restrict__ cursor,
                                 int* __restrict__ perm, int n) {
  int i = blockIdx.x * blockDim.x + threadIdx.x;
  if (i < n) {
    int pos = atomicAdd(&cursor[keys[i]], 1);
    perm[pos] = i;
  }
}

// canonicalize (deterministic): insertion-sort each row segment ascending
__global__ void sort_rows_kernel(const int* __restrict__ offs, int* __restrict__ perm, int nrows) {
  int r = blockIdx.x * blockDim.x + threadIdx.x;
  if (r >= nrows) return;
  int beg = offs[r], end = offs[r + 1];
  for (int i = beg + 1; i < end; ++i) {
    int key = perm[i];
    int j = i - 1;
    while (j >= beg && perm[j] > key) { perm[j + 1] = perm[j]; --j; }
    perm[j + 1] = key;
  }
}

// ============================== init kernels ===============================
__global__ void copy2_kernel(const float* __restrict__ src, float* __restrict__ a,
                             float* __restrict__ b, size_t n) {
  size_t i = (size_t)blockIdx.x * blockDim.x + threadIdx.x;
  if (i < n) { float v = src[i]; a[i] = v; b[i] = v; }
}

__global__ void norm2_kernel(const float* __restrict__ src, float* __restrict__ cur,
                             float* __restrict__ sum, int nrows) {
  int wave = (blockIdx.x * blockDim.x + threadIdx.x) >> 5;
  int lane = threadIdx.x & 31;
  if (wave >= nrows) return;
  const float* x = src + (size_t)wave * DIM;
  float vals[DIM / 32];
  float ss = 0.f;
#pragma unroll
  for (int i = 0; i < DIM / 32; ++i) { vals[i] = x[lane + 32 * i]; ss += vals[i] * vals[i]; }
#pragma unroll
  for (int m = 1; m < 32; m <<= 1) ss += __shfl_xor(ss, m, 32);
  float inv = 1.f / fmaxf(sqrtf(ss), 1e-12f);
  float* c = cur + (size_t)wave * DIM;
  float* s = sum + (size_t)wave * DIM;
#pragma unroll
  for (int i = 0; i < DIM / 32; ++i) { float v = vals[i] * inv; c[lane + 32 * i] = v; s[lane + 32 * i] = v; }
}

__global__ void scale_norm_kernel(float* __restrict__ buf, int nrows, float scale) {
  int wave = (blockIdx.x * blockDim.x + threadIdx.x) >> 5;
  int lane = threadIdx.x & 31;
  if (wave >= nrows) return;
  float* x = buf + (size_t)wave * DIM;
  float vals[DIM / 32];
  float ss = 0.f;
#pragma unroll
  for (int i = 0; i < DIM / 32; ++i) { vals[i] = x[lane + 32 * i] * scale; ss += vals[i] * vals[i]; }
#pragma unroll
  for (int m = 1; m < 32; m <<= 1) ss += __shfl_xor(ss, m, 32);
  float inv = 1.f / fmaxf(sqrtf(ss), 1e-12f);
#pragma unroll
  for (int i = 0; i < DIM / 32; ++i) x[lane + 32 * i] = vals[i] * inv;
}

// ============================== CSR SPMM ===================================
// out[r,:] = sum_{e in row r} vals[perm[e]] * X[gidx[perm[e]], :]
// One wave per output row; register accumulation, no atomics, single store.
// X is L2-resident (51.2MB << 192MB L2); prefetch next gather row.
__global__ void spmm_csr_kernel(const int* __restrict__ offs, const int* __restrict__ perm,
                                const int* __restrict__ gidx, const float* __restrict__ vals,
                                const float* __restrict__ X, float* __restrict__ out, int nrows) {
  int row = (blockIdx.x * blockDim.x + threadIdx.x) >> 5;
  int lane = threadIdx.x & 31;
  if (row >= nrows) return;
  int beg = offs[row], end = offs[row + 1];
  float acc[DIM / 32] = {0.f, 0.f, 0.f, 0.f, 0.f, 0.f, 0.f, 0.f};
  for (int e = beg; e < end; ++e) {
    int id = perm[e];
    if (e + 1 < end) {  // hide gather latency: global_prefetch_b8 of next source row
      int idn = perm[e + 1];
      __builtin_prefetch(X + (size_t)gidx[idn] * DIM + lane, 0, 1);
    }
    float v = vals[id];
    const float* xs = X + (size_t)gidx[id] * DIM;
#pragma unroll
    for (int i = 0; i < DIM / 32; ++i) acc[i] = fmaf(v, xs[lane + 32 * i], acc[i]);
  }
  float* od = out + (size_t)row * DIM;
#pragma unroll
  for (int i = 0; i < DIM / 32; ++i) od[lane + 32 * i] = acc[i];
}

// ============================ WMMA GEMM kernels ============================
// cur[n,:] += relu(T[n,:] @ W^T); sum[n,:] += cur_new.  One wave = 16x64 tile
// (4 accumulators share each A load; 4 independent WMMA chains).
__global__ void gemm_relu_update_kernel(const float* __restrict__ T, const float* __restrict__ W,
                                        float* __restrict__ cur, float* __restrict__ sum, int nrows) {
  int wave = (blockIdx.x * blockDim.x + threadIdx.x) >> 5;
  int lane = threadIdx.x & 31;
  const int NQ = DIM / 64;  // 4 column quads
  int ntiles = (nrows >> 4) * NQ;
  if (wave >= ntiles) return;
  int m0 = (wave / NQ) << 4;
  int nq = (wave % NQ) << 6;
  int h = lane >> 4, nn = lane & 15;
  const float* arow = T + (size_t)(m0 + nn) * DIM + 2 * h;
  const float* brow = W + (size_t)(nq + nn) * DIM + 2 * h;
  v8f c[4];
#pragma unroll
  for (int q = 0; q < 4; ++q) c[q] = (v8f){0.f, 0.f, 0.f, 0.f, 0.f, 0.f, 0.f, 0.f};
#pragma unroll 4
  for (int k0 = 0; k0 < DIM; k0 += 4) {
    v2f a = *(const v2f*)(arow + k0);
#pragma unroll
    for (int q = 0; q < 4; ++q) {
      v2f b = *(const v2f*)(brow + (size_t)q * 16 * DIM + k0);
      c[q] = wmma_f32_16x16x4(a, b, c[q]);
    }
  }
#pragma unroll
  for (int q = 0; q < 4; ++q) {
#pragma unroll
    for (int v = 0; v < 8; ++v) {
      size_t idx = (size_t)(m0 + v + 8 * h) * DIM + nq + q * 16 + nn;
      float r = fmaxf(c[q][v], 0.0f);
      float nv = cur[idx] + r;
      cur[idx] = nv;
      sum[idx] += nv;
    }
  }
}

// rowsum[i] += sum_j exp(dot(A[i],B[j])/TEMP); A,B: [1024,256]; 16x64 per wave
__global__ void logits_rowsum_kernel(const float* __restrict__ A, const float* __restrict__ Bm,
                                     float* __restrict__ rowsum) {
  int wave = (blockIdx.x * blockDim.x + threadIdx.x) >> 5;
  int lane = threadIdx.x & 31;
  const int TQ = BSEL >> 6;  // 16 column quads
  if (wave >= (BSEL >> 4) * TQ) return;
  int m0 = (wave / TQ) << 4;
  int nq = (wave % TQ) << 6;
  int h = lane >> 4, nn = lane & 15;
  const float* arow = A + (size_t)(m0 + nn) * DIM + 2 * h;
  const float* brow = Bm + (size_t)(nq + nn) * DIM + 2 * h;
  v8f c[4];
#pragma unroll
  for (int q = 0; q < 4; ++q) c[q] = (v8f){0.f, 0.f, 0.f, 0.f, 0.f, 0.f, 0.f, 0.f};
#pragma unroll 4
  for (int k0 = 0; k0 < DIM; k0 += 4) {
    v2f a = *(const v2f*)(arow + k0);
#pragma unroll
    for (int q = 0; q < 4; ++q) {
      v2f b = *(const v2f*)(brow + (size_t)q * 16 * DIM + k0);
      c[q] = wmma_f32_16x16x4(a, b, c[q]);
    }
  }
#pragma unroll
  for (int v = 0; v < 8; ++v) {
    float s = 0.f;
#pragma unroll
    for (int q = 0; q < 4; ++q) s += expf(c[q][v] * INV_TEMP);
    s += __shfl_xor(s, 1, 16);
    s += __shfl_xor(s, 2, 16);
    s += __shfl_xor(s, 4, 16);
    s += __shfl_xor(s, 8, 16);
    if (nn == 0) atomicAdd(&rowsum[m0 + v + 8 * h], s);
  }
}

// out2[i,n] += sum_b exp(dot(gs_sel[i], Esf[negs[n*1024+b]])/TEMP); 16x64/wave.
// Flat col c = n*1024+b; 64-col quads never straddle an n boundary.
__global__ void scores_kernel(const float* __restrict__ A, const float* __restrict__ Esf,
                              const int* __restrict__ negs, float* __restrict__ out2) {
  int wave = (blockIdx.x * blockDim.x + threadIdx.x) >> 5;
  int lane = threadIdx.x & 31;
  const int TQ = (NNEGS * BSEL) >> 6;  // 512 column quads
  if (wave >= (BSEL >> 4) * TQ) return;
  int ti = wave / TQ;
  int tq = wave % TQ;
  int m0 = ti << 4;
  int c0 = tq << 6;
  int h = lane >> 4, nn = lane & 15;
  const float* arow = A + (size_t)(m0 + nn) * DIM + 2 * h;
  const float* brow[4];
#pragma unroll
  for (int q = 0; q < 4; ++q) {
    int sid = negs[c0 + q * 16 + nn];
    brow[q] = Esf + (size_t)sid * DIM + 2 * h;
  }
  v8f c[4];
#pragma unroll
  for (int q = 0; q < 4; ++q) c[q] = (v8f){0.f, 0.f, 0.f, 0.f, 0.f, 0.f, 0.f, 0.f};
#pragma unroll 4
  for (int k0 = 0; k0 < DIM; k0 += 4) {
    v2f a = *(const v2f*)(arow + k0);
#pragma unroll
    for (int q = 0; q < 4; ++q) {
      v2f b = *(const v2f*)(brow[q] + k0);
      c[q] = wmma_f32_16x16x4(a, b, c[q]);
    }
  }
  int nidx = c0 >> 10;
#pragma unroll
  for (int v = 0; v < 8; ++v) {
    float s = 0.f;
#pragma unroll
    for (int q = 0; q < 4; ++q) s += expf(c[q][v] * INV_TEMP);
    s += __shfl_xor(s, 1, 16);
    s += __shfl_xor(s, 2, 16);
    s += __shfl_xor(s, 4, 16);
    s += __shfl_xor(s, 8, 16);
    if (nn == 0) atomicAdd(&out2[(size_t)(m0 + v + 8 * h) * NNEGS + nidx], s);
  }
}

// ============================ small epilogues ==============================
__global__ void gather_kernel(const float* __restrict__ src, const int* __restrict__ ids,
                              float* __restrict__ dst, int nrows) {
  int wave = (blockIdx.x * blockDim.x + threadIdx.x) >> 5;
  int lane = threadIdx.x & 31;
  if (wave >= nrows) return;
  int s = ids[wave];
  const float* sp = src + (size_t)s * DIM;
  float* dp = dst + (size_t)wave * DIM;
#pragma unroll
  for (int i = 0; i < DIM / 32; ++i) dp[lane + 32 * i] = sp[lane + 32 * i];
}

__global__ void pos_kernel(const float* __restrict__ g, const float* __restrict__ e,
                           float* __restrict__ acc, int nrows) {
  int wave = (blockIdx.x * blockDim.x + threadIdx.x) >> 5;
  int lane = threadIdx.x & 31;
  if (wave >= nrows) return;
  const float* gp = g + (size_t)wave * DIM;
  const float* ep = e + (size_t)wave * DIM;
  float d = 0.f;
#pragma unroll
  for (int i = 0; i < DIM / 32; ++i) d += gp[lane + 32 * i] * ep[lane + 32 * i];
#pragma unroll
  for (int m = 1; m < 32; m <<= 1) d += __shfl_xor(d, m, 32);
  if (lane == 0) atomicAdd(acc, fminf(fmaxf(d * INV_TEMP, -1.0f), 1.0f));
}

__global__ void sqsum_kernel(const float* __restrict__ a, const float* __restrict__ b,
                             const float* __restrict__ c, const float* __restrict__ d,
                             int n, float* __restrict__ acc) {
  int i = blockIdx.x * blockDim.x + threadIdx.x;
  float s = 0.f;
  if (i < n) s = a[i] * a[i] + b[i] * b[i] + c[i] * c[i] + d[i] * d[i];
#pragma unroll
  for (int m = 1; m < 32; m <<= 1) s += __shfl_xor(s, m, 32);
  __shared__ float sh[8];
  int lane = threadIdx.x & 31, w = threadIdx.x >> 5;
  if (lane == 0) sh[w] = s;
  __syncthreads();
  if (threadIdx.x == 0) {
    float t = 0.f;
    for (int k = 0; k < 8; ++k) t += sh[k];
    atomicAdd(acc, t);
  }
}

__global__ void final_kernel(const float* __restrict__ rowsum1, const float* __restrict__ out2,
                             const float* __restrict__ posacc, const float* __restrict__ sqacc,
                             float* __restrict__ out) {
  float l1 = 0.f, l2 = 0.f;
  for (int i = threadIdx.x; i < BSEL; i += blockDim.x) l1 += logf(rowsum1[i] + 1e-8f);
  for (int i = threadIdx.x; i < BSEL * NNEGS; i += blockDim.x) l2 += logf(out2[i] + 1e-8f);
#pragma unroll
  for (int m = 1; m < 32; m <<= 1) { l1 += __shfl_xor(l1, m, 32); l2 += __shfl_xor(l2, m, 32); }
  __shared__ float s1[8], s2[8];
  int lane = threadIdx.x & 31, w = threadIdx.x >> 5;
  if (lane == 0) { s1[w] = l1; s2[w] = l2; }
  __syncthreads();
  if (threadIdx.x == 0) {
    float t1 = 0.f, t2 = 0.f;
    for (int k = 0; k < 8; ++k) { t1 += s1[k]; t2 += s2[k]; }
    float neg = t1 / (float)BSEL + t2 / (float)(BSEL * NNEGS);
    float pos = posacc[0] / (float)BSEL + posacc[1] / (float)BSEL;
    float cl = (-pos + neg) * 0.2f;
    float reg = 1e-4f * sqacc[0];
    out[0] = cl + reg;
    out[1] = cl;
    out[2] = reg;
  }
}

// ---------------------------------------------------------------------------
extern "C" void kernel_launch(void* const* d_in, const int* in_sizes, int n_in,
                              void* d_out, int out_size, void* d_ws, size_t ws_size,
                              hipStream_t stream) {
  (void)in_sizes; (void)n_in; (void)out_size; (void)ws_size;
  const float* e_j_f   = (const float*)d_in[0];
  const float* e_s_f   = (const float*)d_in[1];
  const float* aug_e_j = (const float*)d_in[2];
  const float* aug_e_s = (const float*)d_in[3];
  const float* val_j   = (const float*)d_in[4];
  const float* val_s   = (const float*)d_in[5];
  const float* W_j     = (const float*)d_in[6];
  const float* W_s     = (const float*)d_in[7];
  const float* W_j_aug = (const float*)d_in[8];
  const float* W_s_aug = (const float*)d_in[9];
  const int* row_j = (const int*)d_in[10];
  const int* col_j = (const int*)d_in[11];
  const int* row_s = (const int*)d_in[12];
  const int* col_s = (const int*)d_in[13];
  const int* j_ids = (const int*)d_in[14];
  const int* s_ids = (const int*)d_in[15];
  const int* negs  = (const int*)d_in[16];

  float* ws = (float*)d_ws;
  const size_t NE = (size_t)NROWS * DIM;  // 12.8M floats
  float* ej  = ws + 0 * NE;  float* es  = ws + 1 * NE;
  float* gj  = ws + 2 * NE;  float* gs  = ws + 3 * NE;
  float* ejs = ws + 4 * NE;  float* ess = ws + 5 * NE;
  float* gjs = ws + 6 * NE;  float* gss = ws + 7 * NE;
  float* tA  = ws + 8 * NE;  float* tB  = ws + 9 * NE;
  float* sel = ws + 10 * NE;
  float* gj_sel = sel + 0 * (size_t)BSEL * DIM;
  float* ej_sel = sel + 1 * (size_t)BSEL * DIM;
  float* gs_sel = sel + 2 * (size_t)BSEL * DIM;
  float* es_sel = sel + 3 * (size_t)BSEL * DIM;
  float* accb   = sel + 4 * (size_t)BSEL * DIM;
  float* rowsum1 = accb;                   // 1024
  float* out2    = accb + BSEL;            // 32768
  float* posacc  = out2 + BSEL * NNEGS;    // 2
  float* sqacc   = posacc + 2;             // 1
  // integer CSR region
  int* ibase  = (int*)(sqacc + 1);
  int* offsJ  = ibase;                 // NROWS+1
  int* offsS  = offsJ + (NROWS + 1);   // NROWS+1
  int* curJ   = offsS + (NROWS + 1);   // NROWS (counts, then cursor)
  int* curS   = curJ + NROWS;          // NROWS
  int* bsum   = curS + NROWS;          // 256 (196 used)
  int* permJ  = bsum + 256;            // NEDGE
  int* permS  = permJ + NEDGE;         // NEDGE

  dim3 blk(256);
  const int edgeBlocks = (NEDGE + 255) / 256;          // 6250
  const int rowBlocks  = (NROWS + 255) / 256;          // 196
  const int waveBlocks = (NROWS * 32 + 255) / 256;     // 6250 (wave-per-row kernels)
  const int gemmBlocks = ((NROWS / 16) * (DIM / 64) * 32 + 255) / 256;  // 1563

  // ---- build CSR for J side (key = row_j, gather = col_j) and S side ------
  hipMemsetAsync(curJ, 0, NROWS * sizeof(int), stream);
  count_kernel<<<edgeBlocks, blk, 0, stream>>>(row_j, curJ, NEDGE);
  scan_block_kernel<<<rowBlocks, blk, 0, stream>>>(curJ, bsum, NROWS);
  scan_spine_kernel<<<1, blk, 0, stream>>>(bsum, rowBlocks);
  scan_final_kernel<<<rowBlocks, blk, 0, stream>>>(curJ, bsum, offsJ, NROWS);
  hipMemcpyAsync(curJ, offsJ, NROWS * sizeof(int), hipMemcpyDeviceToDevice, stream);
  fill_perm_kernel<<<edgeBlocks, blk, 0, stream>>>(row_j, curJ, permJ, NEDGE);
  sort_rows_kernel<<<rowBlocks, blk, 0, stream>>>(offsJ, permJ, NROWS);

  hipMemsetAsync(curS, 0, NROWS * sizeof(int), stream);
  count_kernel<<<edgeBlocks, blk, 0, stream>>>(col_s, curS, NEDGE);
  scan_block_kernel<<<rowBlocks, blk, 0, stream>>>(curS, bsum, NROWS);
  scan_spine_kernel<<<1, blk, 0, stream>>>(bsum, rowBlocks);
  scan_final_kernel<<<rowBlocks, blk, 0, stream>>>(curS, bsum, offsS, NROWS);
  hipMemcpyAsync(curS, offsS, NROWS * sizeof(int), hipMemcpyDeviceToDevice, stream);
  fill_perm_kernel<<<edgeBlocks, blk, 0, stream>>>(col_s, curS, permS, NEDGE);
  sort_rows_kernel<<<rowBlocks, blk, 0, stream>>>(offsS, permS, NROWS);

  // ---- layer 0 state ------------------------------------------------------
  copy2_kernel<<<(unsigned)(NE / 256), blk, 0, stream>>>(e_j_f, ej, ejs, NE);
  copy2_kernel<<<(unsigned)(NE / 256), blk, 0, stream>>>(e_s_f, es, ess, NE);
  norm2_kernel<<<waveBlocks, blk, 0, stream>>>(aug_e_j, gj, gjs, NROWS);
  norm2_kernel<<<waveBlocks, blk, 0, stream>>>(aug_e_s, gs, gss, NROWS);

  for (int l = 0; l < 2; ++l) {
    const float* Wj  = W_j     + (size_t)l * DIM * DIM;
    const float* Ws  = W_s     + (size_t)l * DIM * DIM;
    const float* Wja = W_j_aug + (size_t)l * DIM * DIM;
    const float* Wsa = W_s_aug + (size_t)l * DIM * DIM;
    // E side: both SPMMs read *previous* ej/es, then in-place residual updates
    spmm_csr_kernel<<<waveBlocks, blk, 0, stream>>>(offsJ, permJ, col_j, val_j, es, tA, NROWS);
    spmm_csr_kernel<<<waveBlocks, blk, 0, stream>>>(offsS, permS, row_s, val_s, ej, tB, NROWS);
    gemm_relu_update_kernel<<<gemmBlocks, blk, 0, stream>>>(tA, Wj, ej, ejs, NROWS);
    gemm_relu_update_kernel<<<gemmBlocks, blk, 0, stream>>>(tB, Ws, es, ess, NROWS);
    // G side (stop_gradient is a no-op in forward)
    spmm_csr_kernel<<<waveBlocks, blk, 0, stream>>>(offsJ, permJ, col_j, val_j, gs, tA, NROWS);
    spmm_csr_kernel<<<waveBlocks, blk, 0, stream>>>(offsS, permS, row_s, val_s, gj, tB, NROWS);
    gemm_relu_update_kernel<<<gemmBlocks, blk, 0, stream>>>(tA, Wja, gj, gjs, NROWS);
    gemm_relu_update_kernel<<<gemmBlocks, blk, 0, stream>>>(tB, Wsa, gs, gss, NROWS);
  }

  const float invL = 1.0f / 3.0f;  // 1/(L+1)
  scale_norm_kernel<<<waveBlocks, blk, 0, stream>>>(ejs, NROWS, invL);
  scale_norm_kernel<<<waveBlocks, blk, 0, stream>>>(ess, NROWS, invL);
  scale_norm_kernel<<<waveBlocks, blk, 0, stream>>>(gjs, NROWS, invL);
  scale_norm_kernel<<<waveBlocks, blk, 0, stream>>>(gss, NROWS, invL);

  gather_kernel<<<BSEL / 8, blk, 0, stream>>>(gjs, j_ids, gj_sel, BSEL);
  gather_kernel<<<BSEL / 8, blk, 0, stream>>>(ejs, j_ids, ej_sel, BSEL);
  gather_kernel<<<BSEL / 8, blk, 0, stream>>>(gss, s_ids, gs_sel, BSEL);
  gather_kernel<<<BSEL / 8, blk, 0, stream>>>(ess, s_ids, es_sel, BSEL);

  hipMemsetAsync(accb, 0, (size_t)(BSEL + BSEL * NNEGS + 3) * sizeof(float), stream);
  logits_rowsum_kernel<<<(BSEL / 16) * (BSEL / 64) / 8, blk, 0, stream>>>(gj_sel, ej_sel, rowsum1);
  scores_kernel<<<(BSEL / 16) * ((NNEGS * BSEL) / 64) / 8, blk, 0, stream>>>(gs_sel, ess, negs, out2);
  pos_kernel<<<BSEL / 8, blk, 0, stream>>>(gj_sel, ej_sel, posacc + 0, BSEL);
  pos_kernel<<<BSEL / 8, blk, 0, stream>>>(gs_sel, es_sel, posacc + 1, BSEL);
  sqsum_kernel<<<(2 * DIM * DIM) / 256, blk, 0, stream>>>(W_j, W_s, W_j_aug, W_s_aug,
                                                          2 * DIM * DIM, sqacc);
  final_kernel<<<1, blk, 0, stream>>>(rowsum1, out2, posacc, sqacc, (float*)d_out);
}